// QPLayer_21105469292947
// MI455X (gfx1250) — compile-verified
//
#include <hip/hip_runtime.h>

typedef __attribute__((ext_vector_type(2))) float v2f;
typedef __attribute__((ext_vector_type(8))) float v8f;

#define N_ITER 25
#define SIGMA 0.1f
#define FTB 0.99f
#define NX 64          // nx == m_in
#define MEQ 40
#define NK 104         // KKT dimension (nx + m_eq)
#define KS 105         // KKT row stride: cols 0..103 matrix, col 104 = rhs

__global__ __launch_bounds__(256) void qp_ip_kernel(
    const float*  __restrict__ puzzles,   // [B,64] f32
    const double* __restrict__ Ag,        // [40,64] f64
    const double* __restrict__ lz,        // [64]    f64
    const double* __restrict__ Qg,        // [64,64] f64
    const double* __restrict__ Gg,        // [64,64] f64
    const double* __restrict__ ug,        // [64]    f64
    float* __restrict__ out)              // [B,64] f32
{
    __shared__ float sKKT[NK * KS];       // 43,680 B
    __shared__ float sG[NX * NX];         // 16,384 B
    __shared__ float mult[NK];
    __shared__ float sol[NK];
    __shared__ float red[256];
    __shared__ float xv[NX], yv[MEQ], sv[NX], zv[NX];
    __shared__ float wv[NX], tv[NX], rdv[NX], rpv[MEQ], rsv[NX];
    __shared__ float dsv[NX], dzv[NX], bv[MEQ];

    const int tid  = threadIdx.x;
    const int bid  = blockIdx.x;
    const int lane = tid & 31;
    const int wave = tid >> 5;
    const int l16  = lane & 15;

    // ---- stage G into LDS, init state ----
    for (int i = tid; i < NX * NX; i += 256) sG[i] = (float)Gg[i];
    for (int i = tid; i < NX; i += 256) { xv[i] = 0.f; sv[i] = 1.f; zv[i] = 1.f; }
    for (int i = tid; i < MEQ; i += 256) yv[i] = 0.f;
    __syncthreads();
    // b = A @ exp(log_z0)
    for (int i = tid; i < MEQ; i += 256) {
        float acc = 0.f;
        for (int c = 0; c < NX; ++c)
            acc += (float)Ag[i * NX + c] * __expf((float)lz[c]);
        bv[i] = acc;
    }
    __syncthreads();

    for (int it = 0; it < N_ITER; ++it) {
        // ---- residuals ----
        if (tid < NX) {
            float qx = 0.f, gx = 0.f, aty = 0.f, gtz = 0.f;
            for (int c = 0; c < NX; ++c) {
                qx  += (float)Qg[tid * NX + c] * xv[c];
                gx  += sG[tid * NX + c] * xv[c];
                gtz += sG[c * NX + tid] * zv[c];
            }
            for (int r = 0; r < MEQ; ++r) aty += (float)Ag[r * NX + tid] * yv[r];
            rdv[tid] = qx + (-puzzles[bid * NX + tid]) + aty + gtz;
            rsv[tid] = gx + sv[tid] - (float)ug[tid];
        } else if (tid < NK) {
            int j = tid - NX;
            float ax = 0.f;
            for (int c = 0; c < NX; ++c) ax += (float)Ag[j * NX + c] * xv[c];
            rpv[j] = ax - bv[j];
        }
        __syncthreads();

        // ---- mu = dot(s,z)/m_in ----
        red[tid] = (tid < NX) ? sv[tid] * zv[tid] : 0.f;
        __syncthreads();
        for (int off = 128; off > 0; off >>= 1) {
            if (tid < off) red[tid] += red[tid + off];
            __syncthreads();
        }
        float mu = red[0] / (float)NX;
        __syncthreads();
        if (tid < NX) {
            float s = sv[tid], z = zv[tid];
            wv[tid] = z / s;
            tv[tid] = (SIGMA * mu - z * s + z * rsv[tid]) / s;
        }
        __syncthreads();

        // ---- M = Q + G^T diag(w) G via v_wmma_f32_16x16x4_f32 ----
        // 4x4 grid of 16x16 output tiles; each wave owns two tiles.
        {
            v8f acc0 = {}; v8f acc1 = {};
            const int t0 = wave * 2, t1 = wave * 2 + 1;
            const int m0 = (t0 >> 2) * 16, n0 = (t0 & 3) * 16;
            const int m1 = (t1 >> 2) * 16, n1 = (t1 & 3) * 16;
            const int kh = (lane < 16) ? 0 : 2;   // ISA 16x4 f32 A/B striping
            for (int k0 = 0; k0 < NX; k0 += 4) {
                int ka = k0 + kh;
                float wa = wv[ka], wb = wv[ka + 1];
                v2f a0, b0, a1, b1;
                a0[0] = sG[ka * NX + m0 + l16];       a0[1] = sG[(ka + 1) * NX + m0 + l16];
                b0[0] = wa * sG[ka * NX + n0 + l16];  b0[1] = wb * sG[(ka + 1) * NX + n0 + l16];
                a1[0] = sG[ka * NX + m1 + l16];       a1[1] = sG[(ka + 1) * NX + m1 + l16];
                b1[0] = wa * sG[ka * NX + n1 + l16];  b1[1] = wb * sG[(ka + 1) * NX + n1 + l16];
                acc0 = __builtin_amdgcn_wmma_f32_16x16x4_f32(
                           false, a0, false, b0, (short)0, acc0, false, false);
                acc1 = __builtin_amdgcn_wmma_f32_16x16x4_f32(
                           false, a1, false, b1, (short)0, acc1, false, false);
            }
            // C/D layout: VGPR r -> M=r (lanes 0-15) / M=r+8 (lanes 16-31), N=lane&15
            const int rb = (lane < 16) ? 0 : 8;
            for (int r = 0; r < 8; ++r) {
                int row0 = m0 + rb + r, col0 = n0 + l16;
                sKKT[row0 * KS + col0] = acc0[r] + (float)Qg[row0 * NX + col0];
                int row1 = m1 + rb + r, col1 = n1 + l16;
                sKKT[row1 * KS + col1] = acc1[r] + (float)Qg[row1 * NX + col1];
            }
        }

        // ---- assemble A^T / A / 0 blocks and rhs ----
        for (int idx = tid; idx < NX * MEQ; idx += 256) {
            int i = idx / MEQ, j = idx % MEQ;
            float a = (float)Ag[j * NX + i];
            sKKT[i * KS + (NX + j)] = a;        // top-right A^T
            sKKT[(NX + j) * KS + i] = a;        // bottom-left A
        }
        for (int idx = tid; idx < MEQ * MEQ; idx += 256) {
            int i = idx / MEQ, j = idx % MEQ;
            sKKT[(NX + i) * KS + (NX + j)] = 0.f;
        }
        if (tid < NX) {
            float gt = 0.f;
            for (int k = 0; k < NX; ++k) gt += sG[k * NX + tid] * tv[k];
            sKKT[tid * KS + NK] = -(rdv[tid] + gt);      // rhs_x
        } else if (tid < NK) {
            sKKT[tid * KS + NK] = -rpv[tid - NX];        // -rp
        }
        __syncthreads();

        // ---- Gaussian elimination (in-LDS, cooperative) ----
        for (int k = 0; k < NK - 1; ++k) {
            if (tid > k && tid < NK)
                mult[tid] = sKKT[tid * KS + k] / sKKT[k * KS + k];
            __syncthreads();
            int nrows = NK - 1 - k;
            int ncols = NK - k;                 // cols k+1 .. 104 (incl rhs)
            int total = nrows * ncols;
            for (int idx = tid; idx < total; idx += 256) {
                int i = k + 1 + idx / ncols;
                int j = k + 1 + idx % ncols;
                sKKT[i * KS + j] -= mult[i] * sKKT[k * KS + j];
            }
            __syncthreads();
        }
        // ---- back substitution ----
        for (int k = NK - 1; k >= 0; --k) {
            if (tid == 0) sol[k] = sKKT[k * KS + NK] / sKKT[k * KS + k];
            __syncthreads();
            if (tid < k) sKKT[tid * KS + NK] -= sKKT[tid * KS + k] * sol[k];
            __syncthreads();
        }

        // ---- ds, dz ----
        if (tid < NX) {
            float g = 0.f;
            for (int c = 0; c < NX; ++c) g += sG[tid * NX + c] * sol[c];
            dsv[tid] = -rsv[tid] - g;
            dzv[tid] = tv[tid] + wv[tid] * g;
        }
        __syncthreads();

        // ---- fraction-to-boundary step ----
        float rmin = 3.0e38f;
        if (tid < NX) {
            float ds = dsv[tid], dz = dzv[tid];
            if (ds < 0.f) rmin = fminf(rmin, -sv[tid] / ds);
            if (dz < 0.f) rmin = fminf(rmin, -zv[tid] / dz);
        }
        red[tid] = rmin;
        __syncthreads();
        for (int off = 128; off > 0; off >>= 1) {
            if (tid < off) red[tid] = fminf(red[tid], red[tid + off]);
            __syncthreads();
        }
        float alpha = fminf(1.0f, FTB * red[0]);
        __syncthreads();

        // ---- update ----
        if (tid < NX) {
            xv[tid] += alpha * sol[tid];
            sv[tid] += alpha * dsv[tid];
            zv[tid] += alpha * dzv[tid];
        } else if (tid < NK) {
            yv[tid - NX] += alpha * sol[tid];
        }
        __syncthreads();
    }

    for (int i = tid; i < NX; i += 256) out[bid * NX + i] = xv[i];
}

extern "C" void kernel_launch(void* const* d_in, const int* in_sizes, int n_in,
                              void* d_out, int out_size, void* d_ws, size_t ws_size,
                              hipStream_t stream) {
    (void)n_in; (void)out_size; (void)d_ws; (void)ws_size;
    const float*  puzzles = (const float*) d_in[0];
    const double* A       = (const double*)d_in[1];
    const double* lz      = (const double*)d_in[2];
    const double* Q       = (const double*)d_in[3];
    const double* G       = (const double*)d_in[4];
    const double* u       = (const double*)d_in[5];
    float* out = (float*)d_out;
    int batch = in_sizes[0] / 64;   // puzzles are [B,4,4,4]
    qp_ip_kernel<<<batch, 256, 0, stream>>>(puzzles, A, lz, Q, G, u, out);
}